// GraphSAGE_19524921327629
// MI455X (gfx1250) — compile-verified
//
#include <hip/hip_runtime.h>
#include <hip/hip_bf16.h>

typedef __attribute__((ext_vector_type(2))) float v2f;
typedef __attribute__((ext_vector_type(8))) float v8f;

#define D_FEAT 256  // input feature dim for every layer (d_in = d_h = 256)

// ---------------- zero fill ----------------
__global__ void zero_f32(float* __restrict__ p, int n) {
  int i = blockIdx.x * blockDim.x + threadIdx.x;
  if (i < n) p[i] = 0.0f;
}
__global__ void zero_i32(int* __restrict__ p, int n) {
  int i = blockIdx.x * blockDim.x + threadIdx.x;
  if (i < n) p[i] = 0;
}

// ---------------- edge scatter: agg[dst] += H[src], cnt[dst]++ ----------------
// One thread per (edge, 4-float chunk); 64 chunks cover 256 floats.
__global__ void sage_scatter(const float* __restrict__ H,
                             const int* __restrict__ src,
                             const int* __restrict__ dst,
                             int E,
                             float* __restrict__ agg,
                             int* __restrict__ cnt) {
  long long idx = (long long)blockIdx.x * blockDim.x + threadIdx.x;
  long long total = (long long)E * 64;
  if (idx >= total) return;
  int e = (int)(idx >> 6);
  int c = (int)(idx & 63);
  int s = src[e];
  int d = dst[e];
  const float4 v = ((const float4*)(H + (long long)s * D_FEAT))[c];
  float* p = agg + (long long)d * D_FEAT + c * 4;
  unsafeAtomicAdd(p + 0, v.x);
  unsafeAtomicAdd(p + 1, v.y);
  unsafeAtomicAdd(p + 2, v.z);
  unsafeAtomicAdd(p + 3, v.w);
  if (c == 0) atomicAdd(&cnt[d], 1);
}

// ---------------- inv[m] = 1 / max(cnt[m], 1) ----------------
__global__ void inv_count(const int* __restrict__ cnt, float* __restrict__ inv, int n) {
  int i = blockIdx.x * blockDim.x + threadIdx.x;
  if (i < n) {
    int c = cnt[i];
    inv[i] = 1.0f / (float)(c > 0 ? c : 1);
  }
}

// ---------------- fused dual-GEMM via V_WMMA_F32_16X16X4_F32 ----------------
// out[m,n] = act( sum_k agg[m,k]*inv[m]*Wn[k,n] + sum_k selfA[m,k]*Ws[k,n] + b[n] )
// One wave computes a 16x64 output strip (4 column tiles, v8f acc[4]):
// each A fragment is reused across 4 WMMAs; all loads are unconditional
// (row clamped + 0/1 scale) so the K-loop has no EXEC manipulation.
__global__ void sage_gemm_wmma(const float* __restrict__ agg,
                               const float* __restrict__ inv,
                               const float* __restrict__ selfA,
                               const float* __restrict__ Wn,
                               const float* __restrict__ Ws,
                               const float* __restrict__ bias,
                               float* __restrict__ out,
                               int M, int N, int do_relu) {
  const int K = D_FEAT;
  int wave = threadIdx.x >> 5;
  int lane = threadIdx.x & 31;
  int strips = N >> 6;                       // 64-wide column strips (N % 64 == 0)
  int tilesM = (M + 15) >> 4;
  int tile = blockIdx.x * 8 + wave;          // 8 waves per 256-thread block
  if (tile >= tilesM * strips) return;       // wave-uniform exit (EXEC stays full)
  int tm = tile / strips;
  int tn = tile - tm * strips;
  int row0 = tm << 4;
  int col0 = tn << 6;

  int lrow = row0 + (lane & 15);             // A-matrix row for this lane
  int kb = (lane >> 4) << 1;                 // K sub-offset within the 4-step: 0 or 2
  int nb = col0 + (lane & 15);               // base B/C/D column for this lane

  bool mok = lrow < M;
  int lrc = mok ? lrow : (M - 1);            // clamp: loads always valid
  float invm = mok ? inv[lrc] : 0.0f;        // folds mean-div AND row masking
  float sscl = mok ? 1.0f : 0.0f;            // masks self term for padding rows
  const float* aggRow  = agg   + (long long)lrc * K + kb;
  const float* selfRow = selfA + (long long)lrc * K + kb;
  const float* pWn = Wn + (long long)kb * N + nb;
  const float* pWs = Ws + (long long)kb * N + nb;

  v8f acc[4] = {};
  for (int k0 = 0; k0 < K; k0 += 4) {
    float2 t = *(const float2*)(aggRow);     // contiguous K pair, 8B aligned
    float2 u = *(const float2*)(selfRow);
    v2f a1, a2;
    a1.x = t.x * invm; a1.y = t.y * invm;
    a2.x = u.x * sscl; a2.y = u.y * sscl;
    aggRow += 4; selfRow += 4;
#pragma unroll
    for (int tt = 0; tt < 4; ++tt) {
      v2f b1, b2;
      b1.x = pWn[tt * 16];                   // base + immediate offsets
      b1.y = pWn[N + tt * 16];
      b2.x = pWs[tt * 16];
      b2.y = pWs[N + tt * 16];
      // D = A*B + C, fp32 exact. 8 args: (neg_a, A, neg_b, B, c_mod, C, reuse_a, reuse_b)
      acc[tt] = __builtin_amdgcn_wmma_f32_16x16x4_f32(false, a1, false, b1,
                                                      (short)0, acc[tt], false, false);
      acc[tt] = __builtin_amdgcn_wmma_f32_16x16x4_f32(false, a2, false, b2,
                                                      (short)0, acc[tt], false, false);
    }
    pWn += 4 * N; pWs += 4 * N;
  }

  int mbase = row0 + ((lane >> 4) << 3);     // C/D layout: lanes 16-31 hold M=8..15
#pragma unroll
  for (int tt = 0; tt < 4; ++tt) {
    int n = nb + tt * 16;
    float bn = bias[n];
#pragma unroll
    for (int v = 0; v < 8; ++v) {
      int m = mbase + v;
      if (m < M) {
        float r = acc[tt][v] + bn;
        if (do_relu) r = fmaxf(r, 0.0f);
        out[(long long)m * N + n] = r;
      }
    }
  }
}

// ---------------- host-side orchestration ----------------
static inline int cdiv(long long a, long long b) { return (int)((a + b - 1) / b); }

extern "C" void kernel_launch(void* const* d_in, const int* in_sizes, int n_in,
                              void* d_out, int out_size, void* d_ws, size_t ws_size,
                              hipStream_t stream) {
  const float* x   = (const float*)d_in[0];
  const float* Wn0 = (const float*)d_in[1];
  const float* Ws0 = (const float*)d_in[2];
  const float* b0  = (const float*)d_in[3];
  const float* Wn1 = (const float*)d_in[4];
  const float* Ws1 = (const float*)d_in[5];
  const float* b1  = (const float*)d_in[6];
  const float* Wn2 = (const float*)d_in[7];
  const float* Ws2 = (const float*)d_in[8];
  const float* b2  = (const float*)d_in[9];
  const int* src0  = (const int*)d_in[10];
  const int* dst0  = (const int*)d_in[11];
  const int* src1  = (const int*)d_in[12];
  const int* dst1  = (const int*)d_in[13];
  const int* src2  = (const int*)d_in[14];
  const int* dst2  = (const int*)d_in[15];

  const int N0 = 50000, N1 = 25000, N2 = 12500;
  const int E0 = 500000, E1 = 250000, E2 = 125000;

  // workspace layout (bytes)
  char* ws = (char*)d_ws;
  float* agg = (float*)ws;                                  // 50000*256 f32
  ws += (size_t)N0 * D_FEAT * sizeof(float);
  float* h0 = (float*)ws;                                   // 50000*256 f32
  ws += (size_t)N0 * D_FEAT * sizeof(float);
  float* h1 = (float*)ws;                                   // 25000*256 f32
  ws += (size_t)N1 * D_FEAT * sizeof(float);
  float* inv = (float*)ws;                                  // 50000 f32
  ws += (size_t)N0 * sizeof(float);
  int* cnt = (int*)ws;                                      // 50000 i32

  const int TB = 256;

  // ---- layer 0: x -> h0 (relu) ----
  zero_f32<<<cdiv((long long)N0 * D_FEAT, TB), TB, 0, stream>>>(agg, N0 * D_FEAT);
  zero_i32<<<cdiv(N0, TB), TB, 0, stream>>>(cnt, N0);
  sage_scatter<<<cdiv((long long)E0 * 64, TB), TB, 0, stream>>>(x, src0, dst0, E0, agg, cnt);
  inv_count<<<cdiv(N0, TB), TB, 0, stream>>>(cnt, inv, N0);
  {
    int tiles = ((N0 + 15) / 16) * (256 / 64);
    sage_gemm_wmma<<<cdiv(tiles, 8), TB, 0, stream>>>(agg, inv, x, Wn0, Ws0, b0, h0,
                                                      N0, 256, 1);
  }

  // ---- layer 1: h0 -> h1 (relu) ----
  zero_f32<<<cdiv((long long)N1 * D_FEAT, TB), TB, 0, stream>>>(agg, N1 * D_FEAT);
  zero_i32<<<cdiv(N1, TB), TB, 0, stream>>>(cnt, N1);
  sage_scatter<<<cdiv((long long)E1 * 64, TB), TB, 0, stream>>>(h0, src1, dst1, E1, agg, cnt);
  inv_count<<<cdiv(N1, TB), TB, 0, stream>>>(cnt, inv, N1);
  {
    int tiles = ((N1 + 15) / 16) * (256 / 64);
    sage_gemm_wmma<<<cdiv(tiles, 8), TB, 0, stream>>>(agg, inv, h0, Wn1, Ws1, b1, h1,
                                                      N1, 256, 1);
  }

  // ---- layer 2: h1 -> out (no relu, N=128) ----
  zero_f32<<<cdiv((long long)N2 * D_FEAT, TB), TB, 0, stream>>>(agg, N2 * D_FEAT);
  zero_i32<<<cdiv(N2, TB), TB, 0, stream>>>(cnt, N2);
  sage_scatter<<<cdiv((long long)E2 * 64, TB), TB, 0, stream>>>(h1, src2, dst2, E2, agg, cnt);
  inv_count<<<cdiv(N2, TB), TB, 0, stream>>>(cnt, inv, N2);
  {
    int tiles = ((N2 + 15) / 16) * (128 / 64);
    sage_gemm_wmma<<<cdiv(tiles, 8), TB, 0, stream>>>(agg, inv, h1, Wn2, Ws2, b2,
                                                      (float*)d_out, N2, 128, 0);
  }
}